// GCN_20882130993434
// MI455X (gfx1250) — compile-verified
//
#include <hip/hip_runtime.h>
#include <math.h>

// ---------------- problem constants (match reference) ----------------
#define N_NODES    100000
#define N_EDGES    1600000
#define IN_CH      128
#define HID_CH     128
#define OUT_CH     64
#define NUM_GRAPHS 512

typedef __attribute__((ext_vector_type(2))) float v2f;
typedef __attribute__((ext_vector_type(8))) float v8f;

// =====================================================================
// WMMA f32 GEMM:  C[nrows, NCOL] = A[nrows, 128] @ W[128, NCOL] (+bias)
// Full fp32 via V_WMMA_F32_16X16X4_F32 (keeps reference precision).
// Block = 256 threads = 8 waves; each wave computes a 16-row strip.
// W (<=64KB) staged in LDS once per block.
// =====================================================================
template <int NCOL, bool ADD_BIAS>
__global__ __launch_bounds__(256)
void gemm_wmma_f32(const float* __restrict__ A, const float* __restrict__ W,
                   const float* __restrict__ bias, float* __restrict__ C,
                   int nrows)
{
    __shared__ float sW[128 * NCOL];

    const int tid = threadIdx.x;

    // cooperative 128xNCOL weight stage (float4 -> ds_store_b128)
    {
        const float4* W4  = (const float4*)W;
        float4*       sW4 = (float4*)sW;
        #pragma unroll 4
        for (int i = tid; i < (128 * NCOL) / 4; i += 256) sW4[i] = W4[i];
    }
    __syncthreads();

    const int wave = tid >> 5;
    const int lane = tid & 31;
    const int half = lane >> 4;   // 0: lanes 0-15, 1: lanes 16-31
    const int l16  = lane & 15;
    const int rowBase = blockIdx.x * 128 + wave * 16;

    v8f acc[NCOL / 16];
    #pragma unroll
    for (int t = 0; t < NCOL / 16; ++t) acc[t] = (v8f){};

    // A-fragment row for this lane (clamped so EXEC stays all-1s for WMMA)
    int r = rowBase + l16;
    if (r > nrows - 1) r = nrows - 1;
    const float* Arow = A + (size_t)r * 128;

    for (int k0 = 0; k0 < 128; k0 += 4) {
        // A 16x4 f32 layout: v0 = K0 (lo lanes)/K2 (hi lanes), v1 = K1/K3
        // -> per-lane contiguous 8B load
        v2f a = *(const v2f*)(Arow + k0 + 2 * half);
        #pragma unroll
        for (int t = 0; t < NCOL / 16; ++t) {
            v2f b;
            b.x = sW[(k0 + 2 * half + 0) * NCOL + t * 16 + l16];
            b.y = sW[(k0 + 2 * half + 1) * NCOL + t * 16 + l16];
            acc[t] = __builtin_amdgcn_wmma_f32_16x16x4_f32(
                false, a, false, b, (short)0, acc[t], false, false);
        }
    }

    // C/D 16x16 f32 layout: VGPR v -> row = v + 8*half, col = l16
    #pragma unroll
    for (int t = 0; t < NCOL / 16; ++t) {
        #pragma unroll
        for (int v = 0; v < 8; ++v) {
            int row = rowBase + v + 8 * half;
            if (row < nrows) {
                float o = acc[t][v];
                if (ADD_BIAS) o += bias[t * 16 + l16];
                C[(size_t)row * NCOL + t * 16 + l16] = o;
            }
        }
    }
}

// ---------------- degree / normalization ----------------
__global__ void k_fill(float* p, float v, int n) {
    int i = blockIdx.x * blockDim.x + threadIdx.x;
    if (i < n) p[i] = v;
}
__global__ void k_deg_count(const int* __restrict__ dst, float* __restrict__ deg, int nE) {
    int e = blockIdx.x * blockDim.x + threadIdx.x;
    if (e < nE) atomicAdd(&deg[dst[e]], 1.0f);
}
__global__ void k_rsqrt_inplace(float* p, int n) {
    int i = blockIdx.x * blockDim.x + threadIdx.x;
    if (i < n) p[i] = rsqrtf(p[i]);
}

// agg[n,c] = h[n,c] * dinv[n]^2   (self-loop term; also zero-initializes agg)
__global__ __launch_bounds__(256)
void k_init_agg(const float* __restrict__ h, const float* __restrict__ dinv,
                float* __restrict__ agg)
{
    int gid  = blockIdx.x * 256 + threadIdx.x;          // one float4 per thread
    int node = gid >> 5;                                // 32 threads * 4ch = 128ch
    if (node >= N_NODES) return;
    float d  = dinv[node];
    float s  = d * d;
    float4 hv = *(const float4*)(h + (size_t)node * 128 + (gid & 31) * 4);
    float4 o  = {hv.x * s, hv.y * s, hv.z * s, hv.w * s};
    *(float4*)(agg + (size_t)node * 128 + (gid & 31) * 4) = o;
}

// agg[dst] += h[src] * dinv[src]*dinv[dst]   (1 wave per edge, f32 L2 atomics)
__global__ __launch_bounds__(256)
void k_scatter(const float* __restrict__ h, const float* __restrict__ dinv,
               const int* __restrict__ src, const int* __restrict__ dst,
               float* __restrict__ agg, int nE)
{
    int gid  = blockIdx.x * 256 + threadIdx.x;
    int e    = gid >> 5;
    int lane = gid & 31;
    if (e >= nE) return;
    int s = src[e], d = dst[e];
    float norm = dinv[s] * dinv[d];
    float4 hv = *(const float4*)(h + (size_t)s * 128 + lane * 4);
    float* out = agg + (size_t)d * 128 + lane * 4;
    atomicAdd(out + 0, hv.x * norm);
    atomicAdd(out + 1, hv.y * norm);
    atomicAdd(out + 2, hv.z * norm);
    atomicAdd(out + 3, hv.w * norm);
}

// out = (relu?)(agg + b[c])
template <bool RELU>
__global__ __launch_bounds__(256)
void k_finalize(const float* __restrict__ agg, const float* __restrict__ b,
                float* __restrict__ out)
{
    int gid  = blockIdx.x * 256 + threadIdx.x;          // one float4 per thread
    int node = gid >> 5;
    if (node >= N_NODES) return;
    int c = (gid & 31) * 4;
    float4 a  = *(const float4*)(agg + (size_t)node * 128 + c);
    float4 bb = *(const float4*)(b + c);
    float4 o  = {a.x + bb.x, a.y + bb.y, a.z + bb.z, a.w + bb.w};
    if (RELU) {
        o.x = fmaxf(o.x, 0.f); o.y = fmaxf(o.y, 0.f);
        o.z = fmaxf(o.z, 0.f); o.w = fmaxf(o.w, 0.f);
    }
    *(float4*)(out + (size_t)node * 128 + c) = o;
}

// ---------------- mean pool ----------------
__global__ __launch_bounds__(256)
void k_pool_accum(const float* __restrict__ h, const int* __restrict__ batch,
                  float* __restrict__ sums, float* __restrict__ cnts)
{
    int gid  = blockIdx.x * 256 + threadIdx.x;
    int node = gid >> 5;
    int lane = gid & 31;
    if (node >= N_NODES) return;
    int g = batch[node];
    float4 hv = *(const float4*)(h + (size_t)node * 128 + lane * 4);
    float* out = sums + (size_t)g * 128 + lane * 4;
    atomicAdd(out + 0, hv.x);
    atomicAdd(out + 1, hv.y);
    atomicAdd(out + 2, hv.z);
    atomicAdd(out + 3, hv.w);
    if (lane == 0) atomicAdd(&cnts[g], 1.0f);
}

__global__ void k_pool_div(const float* __restrict__ sums, const float* __restrict__ cnts,
                           float* __restrict__ pooled)
{
    int i = blockIdx.x * blockDim.x + threadIdx.x;
    if (i < NUM_GRAPHS * 128) pooled[i] = sums[i] / fmaxf(cnts[i >> 7], 1.0f);
}

// =====================================================================
extern "C" void kernel_launch(void* const* d_in, const int* in_sizes, int n_in,
                              void* d_out, int out_size, void* d_ws, size_t ws_size,
                              hipStream_t stream)
{
    const float* x   = (const float*)d_in[0];
    const float* W1  = (const float*)d_in[1];
    const float* b1  = (const float*)d_in[2];
    const float* W2  = (const float*)d_in[3];
    const float* b2  = (const float*)d_in[4];
    const float* Wfc = (const float*)d_in[5];
    const float* bfc = (const float*)d_in[6];
    const int*   ei  = (const int*)d_in[7];      // [2, E] flattened
    const int*   bat = (const int*)d_in[8];
    const int* src = ei;
    const int* dst = ei + N_EDGES;
    float* out = (float*)d_out;                  // [512, 64]

    // workspace layout (floats); total ~154.5 MB (fits the 192 MB L2)
    const size_t SZ_H = (size_t)N_NODES * HID_CH;        // 12.8M floats
    float* hA     = (float*)d_ws;
    float* hB     = hA  + SZ_H;
    float* agg    = hB  + SZ_H;
    float* dinv   = agg + SZ_H;                          // N_NODES
    float* sums   = dinv + N_NODES;                      // 512*128
    float* cnts   = sums + NUM_GRAPHS * 128;             // 512
    float* pooled = cnts + NUM_GRAPHS;                   // 512*128

    const int gemmBlocks = (N_NODES + 127) / 128;        // 782
    const int nodeVecBlk = (N_NODES * 32 + 255) / 256;   // 12500 (float4/thread)
    const int edgeWaveBlk = (N_EDGES * 32 + 255) / 256;  // 200000 (wave/edge)

    // degree -> dinv (shared by both conv layers)
    k_fill<<<(N_NODES + 255) / 256, 256, 0, stream>>>(dinv, 1.0f, N_NODES);
    k_deg_count<<<(N_EDGES + 255) / 256, 256, 0, stream>>>(dst, dinv, N_EDGES);
    k_rsqrt_inplace<<<(N_NODES + 255) / 256, 256, 0, stream>>>(dinv, N_NODES);

    // ---- layer 1 ----
    gemm_wmma_f32<128, false><<<gemmBlocks, 256, 0, stream>>>(x, W1, nullptr, hA, N_NODES);
    k_init_agg<<<nodeVecBlk, 256, 0, stream>>>(hA, dinv, agg);
    k_scatter<<<edgeWaveBlk, 256, 0, stream>>>(hA, dinv, src, dst, agg, N_EDGES);
    k_finalize<true><<<nodeVecBlk, 256, 0, stream>>>(agg, b1, hB);

    // ---- layer 2 ----
    gemm_wmma_f32<128, false><<<gemmBlocks, 256, 0, stream>>>(hB, W2, nullptr, hA, N_NODES);
    k_init_agg<<<nodeVecBlk, 256, 0, stream>>>(hA, dinv, agg);
    k_scatter<<<edgeWaveBlk, 256, 0, stream>>>(hA, dinv, src, dst, agg, N_EDGES);
    k_finalize<false><<<nodeVecBlk, 256, 0, stream>>>(agg, b2, hB);

    // ---- global mean pool ----
    k_fill<<<(NUM_GRAPHS * 128 + 255) / 256, 256, 0, stream>>>(sums, 0.0f, NUM_GRAPHS * 128);
    k_fill<<<(NUM_GRAPHS + 255) / 256, 256, 0, stream>>>(cnts, 0.0f, NUM_GRAPHS);
    k_pool_accum<<<nodeVecBlk, 256, 0, stream>>>(hB, bat, sums, cnts);
    k_pool_div<<<(NUM_GRAPHS * 128 + 255) / 256, 256, 0, stream>>>(sums, cnts, pooled);

    // ---- FC: [512,128] @ [128,64] + bfc ----
    gemm_wmma_f32<64, true><<<NUM_GRAPHS / 128, 256, 0, stream>>>(pooled, Wfc, bfc, out, NUM_GRAPHS);
}